// GraphSAGE_70076686401960
// MI455X (gfx1250) — compile-verified
//
#include <hip/hip_runtime.h>
#include <hip/hip_bf16.h>

// ---------------------------------------------------------------------------
// GraphSAGE (2x SAGEConv mean-aggr + FC) for MI455X / gfx1250, wave32.
// All-f32 path: workload is L2-atomic bound (edge scatter), compute is ~2
// GFLOP total, so we use V_WMMA_F32_16X16X4_F32 for the dense projections
// to keep full precision while exercising the CDNA5 matrix pipe.
// ---------------------------------------------------------------------------

typedef __attribute__((ext_vector_type(2))) float v2f;
typedef __attribute__((ext_vector_type(8))) float v8f;

static __device__ __forceinline__ v8f wmma_f32(v2f a, v2f b, v8f c) {
    // D = A(16x4,f32) * B(4x16,f32) + C(16x16,f32)
    return __builtin_amdgcn_wmma_f32_16x16x4_f32(
        /*neg_a=*/false, a, /*neg_b=*/false, b,
        /*c_mod=*/(short)0, c, /*reuse_a=*/false, /*reuse_b=*/false);
}

// ------------------------------ zero scratch -------------------------------
__global__ void zero_f32_kernel(float* __restrict__ p, unsigned long long n) {
    unsigned long long i = (unsigned long long)blockIdx.x * blockDim.x + threadIdx.x;
    unsigned long long stride = (unsigned long long)gridDim.x * blockDim.x;
    for (; i < n; i += stride) p[i] = 0.0f;
}

// ------------------- edge scatter, layer 1 (128 channels) ------------------
// thread i handles edge e = i/32, channel group c4 = (i%32)*4.
__global__ void scatter128_kernel(const int* __restrict__ src,
                                  const int* __restrict__ dst,
                                  const float* __restrict__ x,
                                  float* __restrict__ agg,
                                  float* __restrict__ deg,
                                  unsigned long long total) {
    unsigned long long i = (unsigned long long)blockIdx.x * blockDim.x + threadIdx.x;
    if (i >= total) return;
    const unsigned long long e  = i >> 5;
    const unsigned          c4 = ((unsigned)i & 31u) << 2;
    const int s = src[e];
    const int d = dst[e];
    const float4 v = *(const float4*)(x + (unsigned long long)s * 128u + c4);
    float* a = agg + (unsigned long long)d * 128u + c4;
    atomicAdd(a + 0, v.x);
    atomicAdd(a + 1, v.y);
    atomicAdd(a + 2, v.z);
    atomicAdd(a + 3, v.w);
    if (c4 == 0) atomicAdd(deg + d, 1.0f);
}

// ------------------- edge scatter, layer 2 (32 channels) -------------------
__global__ void scatter32_kernel(const int* __restrict__ src,
                                 const int* __restrict__ dst,
                                 const float* __restrict__ h,
                                 float* __restrict__ agg,
                                 unsigned long long total) {
    unsigned long long i = (unsigned long long)blockIdx.x * blockDim.x + threadIdx.x;
    if (i >= total) return;
    const unsigned long long e  = i >> 3;
    const unsigned          c4 = ((unsigned)i & 7u) << 2;
    const int s = src[e];
    const int d = dst[e];
    const float4 v = *(const float4*)(h + (unsigned long long)s * 32u + c4);
    float* a = agg + (unsigned long long)d * 32u + c4;
    atomicAdd(a + 0, v.x);
    atomicAdd(a + 1, v.y);
    atomicAdd(a + 2, v.z);
    atomicAdd(a + 3, v.w);
}

// --------------------- layer 1: WMMA node kernel ---------------------------
// out[16n,32] = relu( (agg/deg)[16n,128] @ Wl[128,32] + x[16n,128] @ Wr[128,32] + b )
// One wave per 16-node tile. K=128 in steps of 4; two 16-col N-tiles.
__global__ __launch_bounds__(128)
void sage1_wmma_kernel(const float* __restrict__ x,
                       const float* __restrict__ agg,
                       const float* __restrict__ deg,
                       const float* __restrict__ Wl,
                       const float* __restrict__ Wr,
                       const float* __restrict__ bias,
                       float* __restrict__ out,
                       int ntiles) {
    const int lane = threadIdx.x & 31;
    const int tile = blockIdx.x * (blockDim.x >> 5) + (threadIdx.x >> 5);
    if (tile >= ntiles) return;              // wave-uniform: EXEC all-1s for WMMA

    const int m    = lane & 15;              // node-in-tile (A rows) / column n (B,C,D)
    const int half = lane >> 4;              // 0: K={0,1}, 1: K={2,3}
    const int kk   = half << 1;
    const int node = tile * 16 + m;

    const float dinv = 1.0f / fmaxf(deg[node], 1.0f);
    const float* xrow = x   + (unsigned long long)node * 128u;
    const float* arow = agg + (unsigned long long)node * 128u;

    v8f acc0, acc1;
    const float b0 = bias[m];
    const float b1v = bias[16 + m];
#pragma unroll
    for (int i = 0; i < 8; ++i) { acc0[i] = b0; acc1[i] = b1v; }

#pragma unroll 4
    for (int k0 = 0; k0 < 128; k0 += 4) {
        v2f aM, aX;
        aM.x = arow[k0 + kk]     * dinv;
        aM.y = arow[k0 + kk + 1] * dinv;
        aX.x = xrow[k0 + kk];
        aX.y = xrow[k0 + kk + 1];

        const float* wl = Wl + (unsigned long long)(k0 + kk) * 32u;
        const float* wr = Wr + (unsigned long long)(k0 + kk) * 32u;
        v2f bl0, bl1, br0, br1;
        bl0.x = wl[m];          bl0.y = wl[32 + m];
        bl1.x = wl[16 + m];     bl1.y = wl[48 + m];
        br0.x = wr[m];          br0.y = wr[32 + m];
        br1.x = wr[16 + m];     br1.y = wr[48 + m];

        acc0 = wmma_f32(aM, bl0, acc0);
        acc0 = wmma_f32(aX, br0, acc0);
        acc1 = wmma_f32(aM, bl1, acc1);
        acc1 = wmma_f32(aX, br1, acc1);
    }

    // C/D layout: VGPR v holds row M = v + half*8, column = m (acc1: +16)
#pragma unroll
    for (int v = 0; v < 8; ++v) {
        const int row = tile * 16 + v + half * 8;
        out[(unsigned long long)row * 32u + m]      = fmaxf(acc0[v], 0.0f);
        out[(unsigned long long)row * 32u + 16 + m] = fmaxf(acc1[v], 0.0f);
    }
}

// --------------------- layer 2: WMMA node kernel ---------------------------
// out[16n,16] = relu( (agg/deg)[16n,32] @ Wl[32,16] + h[16n,32] @ Wr[32,16] + b )
__global__ __launch_bounds__(128)
void sage2_wmma_kernel(const float* __restrict__ h,
                       const float* __restrict__ agg,
                       const float* __restrict__ deg,
                       const float* __restrict__ Wl,
                       const float* __restrict__ Wr,
                       const float* __restrict__ bias,
                       float* __restrict__ out,
                       int ntiles) {
    const int lane = threadIdx.x & 31;
    const int tile = blockIdx.x * (blockDim.x >> 5) + (threadIdx.x >> 5);
    if (tile >= ntiles) return;

    const int m    = lane & 15;
    const int half = lane >> 4;
    const int kk   = half << 1;
    const int node = tile * 16 + m;

    const float dinv = 1.0f / fmaxf(deg[node], 1.0f);
    const float* hrow = h   + (unsigned long long)node * 32u;
    const float* arow = agg + (unsigned long long)node * 32u;

    v8f acc;
    const float b0 = bias[m];
#pragma unroll
    for (int i = 0; i < 8; ++i) acc[i] = b0;

#pragma unroll
    for (int k0 = 0; k0 < 32; k0 += 4) {
        v2f aM, aX;
        aM.x = arow[k0 + kk]     * dinv;
        aM.y = arow[k0 + kk + 1] * dinv;
        aX.x = hrow[k0 + kk];
        aX.y = hrow[k0 + kk + 1];

        const float* wl = Wl + (unsigned long long)(k0 + kk) * 16u;
        const float* wr = Wr + (unsigned long long)(k0 + kk) * 16u;
        v2f bl, br;
        bl.x = wl[m];   bl.y = wl[16 + m];
        br.x = wr[m];   br.y = wr[16 + m];

        acc = wmma_f32(aM, bl, acc);
        acc = wmma_f32(aX, br, acc);
    }

#pragma unroll
    for (int v = 0; v < 8; ++v) {
        const int row = tile * 16 + v + half * 8;
        out[(unsigned long long)row * 16u + m] = fmaxf(acc[v], 0.0f);
    }
}

// ------------------------------ final FC -----------------------------------
__global__ void fc_kernel(const float* __restrict__ h2,
                          const float* __restrict__ Wfc,
                          const float* __restrict__ bfc,
                          float* __restrict__ out, int n_nodes) {
    const int n = blockIdx.x * blockDim.x + threadIdx.x;
    if (n >= n_nodes) return;
    float o0 = bfc[0], o1 = bfc[1];
    const float* row = h2 + (unsigned long long)n * 16u;
#pragma unroll
    for (int k = 0; k < 16; ++k) {
        const float v = row[k];
        o0 = fmaf(v, Wfc[k * 2 + 0], o0);
        o1 = fmaf(v, Wfc[k * 2 + 1], o1);
    }
    out[(unsigned long long)n * 2u + 0] = o0;
    out[(unsigned long long)n * 2u + 1] = o1;
}

// ---------------------------------------------------------------------------
extern "C" void kernel_launch(void* const* d_in, const int* in_sizes, int n_in,
                              void* d_out, int out_size, void* d_ws, size_t ws_size,
                              hipStream_t stream) {
    (void)n_in; (void)out_size; (void)ws_size;

    const float* x    = (const float*)d_in[0];
    const int*   ei   = (const int*)d_in[1];     // edge_index flat [2, E]
    const float* W1l  = (const float*)d_in[2];
    const float* b1   = (const float*)d_in[3];
    const float* W1r  = (const float*)d_in[4];
    const float* W2l  = (const float*)d_in[5];
    const float* b2   = (const float*)d_in[6];
    const float* W2r  = (const float*)d_in[7];
    const float* Wfc  = (const float*)d_in[8];
    const float* bfc  = (const float*)d_in[9];
    float*       out  = (float*)d_out;

    const int N = in_sizes[0] / 128;             // 100000
    const int E = in_sizes[1] / 2;               // 1600000
    const int* src = ei;
    const int* dst = ei + E;

    // Workspace layout (floats): [agg1 N*128][deg N][agg2 N*32][h1 N*32][h2 N*16]
    float* ws   = (float*)d_ws;
    float* agg1 = ws;
    float* deg  = agg1 + (unsigned long long)N * 128u;
    float* agg2 = deg  + (unsigned long long)N;
    float* h1   = agg2 + (unsigned long long)N * 32u;
    float* h2   = h1   + (unsigned long long)N * 32u;

    // 1) zero [agg1 | deg | agg2] (contiguous) — rerun every call for replay safety
    {
        const unsigned long long nz = (unsigned long long)N * (128u + 1u + 32u);
        zero_f32_kernel<<<8192, 256, 0, stream>>>(agg1, nz);
    }

    // 2) layer-1 edge scatter (+degree)
    {
        const unsigned long long total = (unsigned long long)E * 32u;
        const unsigned blocks = (unsigned)((total + 255u) / 256u);
        scatter128_kernel<<<blocks, 256, 0, stream>>>(src, dst, x, agg1, deg, total);
    }

    // 3) layer-1 dense projection (WMMA f32) + ReLU
    {
        const int ntiles = (N + 15) / 16;        // N divisible by 16 here
        const int blocks = (ntiles + 3) / 4;     // 4 waves / block
        sage1_wmma_kernel<<<blocks, 128, 0, stream>>>(x, agg1, deg, W1l, W1r, b1, h1, ntiles);
    }

    // 4) layer-2 edge scatter
    {
        const unsigned long long total = (unsigned long long)E * 8u;
        const unsigned blocks = (unsigned)((total + 255u) / 256u);
        scatter32_kernel<<<blocks, 256, 0, stream>>>(src, dst, h1, agg2, total);
    }

    // 5) layer-2 dense projection (WMMA f32) + ReLU
    {
        const int ntiles = (N + 15) / 16;
        const int blocks = (ntiles + 3) / 4;
        sage2_wmma_kernel<<<blocks, 128, 0, stream>>>(h1, agg2, deg, W2l, W2r, b2, h2, ntiles);
    }

    // 6) final FC [16 -> 2]
    {
        const int blocks = (N + 255) / 256;
        fc_kernel<<<blocks, 256, 0, stream>>>(h2, Wfc, bfc, out, N);
    }
}